// SAGEEncoder_31756988186713
// MI455X (gfx1250) — compile-verified
//
#include <hip/hip_runtime.h>
#include <hip/hip_bf16.h>

#define D 128

typedef __attribute__((ext_vector_type(2))) float v2f;
typedef __attribute__((ext_vector_type(8))) float v8f;

// ---------------------------------------------------------------- utilities
__global__ void fill_zero_kernel(float* __restrict__ p, long long n) {
  long long i = (long long)blockIdx.x * blockDim.x + threadIdx.x;
  long long stride = (long long)gridDim.x * blockDim.x;
  for (; i < n; i += stride) p[i] = 0.0f;
}

__global__ void deg_count_kernel(const int* __restrict__ dst,
                                 float* __restrict__ deg, int ne) {
  int e = blockIdx.x * blockDim.x + threadIdx.x;
  if (e < ne) atomicAdd(&deg[dst[e]], 1.0f);
}

__global__ void deg_inv_kernel(float* __restrict__ deg, int n) {
  int i = blockIdx.x * blockDim.x + threadIdx.x;
  if (i < n) deg[i] = 1.0f / fmaxf(deg[i], 1.0f);
}

// --------------------------------------------------- edge scatter (mean num)
// one edge handled by 32 threads; each thread adds 4 contiguous features.
__global__ void scatter_add_kernel(const float* __restrict__ x,
                                   const int* __restrict__ src,
                                   const int* __restrict__ dst,
                                   float* __restrict__ agg, int ne) {
  int t = blockIdx.x * blockDim.x + threadIdx.x;
  int e = t >> 5;
  if (e >= ne) return;
  int c = (t & 31) << 2;
  int s = src[e];
  int d = dst[e];
  const float4 v = *(const float4*)(x + (size_t)s * D + c);
  float* a = agg + (size_t)d * D + c;
  atomicAdd(a + 0, v.x);
  atomicAdd(a + 1, v.y);
  atomicAdd(a + 2, v.z);
  atomicAdd(a + 3, v.w);
}

// -------------------------------------------------- fused SAGE layer (WMMA)
// out = relu( (agg*invdeg) @ Wl^T + bl + x @ Wr^T )
// block: 16 node rows x 128 cols; 8 waves, wave w owns columns [16w,16w+16).
// A operands staged in LDS (stride 132 -> conflict-free ds_load_b64).
__launch_bounds__(256, 2)
__global__ void sage_layer_kernel(const float* __restrict__ x,
                                  const float* __restrict__ agg,
                                  const float* __restrict__ invdeg,
                                  const float* __restrict__ Wl,
                                  const float* __restrict__ bl,
                                  const float* __restrict__ Wr,
                                  float* __restrict__ out) {
  __shared__ float sA[16 * 132];  // inv_deg-scaled aggregate tile
  __shared__ float sX[16 * 132];  // residual x tile

  const int tid = threadIdx.x;

  // ---- cooperative staging: 256 threads x 8 floats per buffer -------------
  {
    int row = tid >> 4;          // 0..15
    int col = (tid & 15) << 3;   // 0,8,...,120
    size_t g = (size_t)blockIdx.x * 16 + row;
    float s = invdeg[g];
    const float4* pa = (const float4*)(agg + g * D + col);
    const float4* px = (const float4*)(x + g * D + col);
    float4 a0 = pa[0], a1 = pa[1];
    float4 x0 = px[0], x1 = px[1];
    a0.x *= s; a0.y *= s; a0.z *= s; a0.w *= s;
    a1.x *= s; a1.y *= s; a1.z *= s; a1.w *= s;
    *(float4*)&sA[row * 132 + col + 0] = a0;
    *(float4*)&sA[row * 132 + col + 4] = a1;
    *(float4*)&sX[row * 132 + col + 0] = x0;
    *(float4*)&sX[row * 132 + col + 4] = x1;
  }
  __syncthreads();

  const int lane = tid & 31;
  const int wave = tid >> 5;          // column tile 0..7
  const int r16  = lane & 15;
  const int h    = lane >> 4;         // lane-half selects K pair / M half
  const int n_col = wave * 16 + r16;  // output column (= Wl/Wr row)

  const float* wl = Wl + (size_t)n_col * D;  // B[k][n] = Wl[n][k] (x @ Wl^T)
  const float* wr = Wr + (size_t)n_col * D;
  __builtin_prefetch(wl, 0, 0);
  __builtin_prefetch(wr, 0, 0);

  v8f c = {0.f, 0.f, 0.f, 0.f, 0.f, 0.f, 0.f, 0.f};

  // K loop: 128 / 4 = 32 steps, two f32 WMMAs per step (lin_l + lin_r fused)
  for (int k = 0; k < D; k += 4) {
    const int kk = k + 2 * h;  // per ISA 16x4 f32 A layout: K = v + 2h (+k)
    v2f aA = *(const v2f*)&sA[r16 * 132 + kk];
    v2f aX = *(const v2f*)&sX[r16 * 132 + kk];
    v2f bL = *(const v2f*)&wl[kk];
    v2f bR = *(const v2f*)&wr[kk];
    c = __builtin_amdgcn_wmma_f32_16x16x4_f32(false, aA, false, bL,
                                              (short)0, c, false, false);
    c = __builtin_amdgcn_wmma_f32_16x16x4_f32(false, aX, false, bR,
                                              (short)0, c, false, false);
  }

  // ---- epilogue: bias + ReLU; C/D layout: M = r + 8h, N = lane%16 ---------
  const float blv = bl[n_col];
#pragma unroll
  for (int r = 0; r < 8; ++r) {
    size_t m = (size_t)blockIdx.x * 16 + r + 8 * h;
    out[m * D + n_col] = fmaxf(c[r] + blv, 0.0f);
  }
}

// ------------------------------------------------------------------- driver
extern "C" void kernel_launch(void* const* d_in, const int* in_sizes, int n_in,
                              void* d_out, int out_size, void* d_ws, size_t ws_size,
                              hipStream_t stream) {
  const float* x0 = (const float*)d_in[0];
  const int*   ei = (const int*)d_in[1];
  const float* Wl = (const float*)d_in[2];
  const float* bl = (const float*)d_in[3];
  const float* Wr = (const float*)d_in[4];

  const int n_nodes  = in_sizes[0] / D;
  const int n_edges  = in_sizes[1] / 2;
  const int n_layers = in_sizes[2] / (D * D);

  const int* src = ei;
  const int* dst = ei + n_edges;

  float* agg    = (float*)d_ws;                       // [N, D]
  float* bufA   = agg + (size_t)n_nodes * D;          // [N, D] ping-pong
  float* invdeg = bufA + (size_t)n_nodes * D;         // [N]
  float* outF   = (float*)d_out;

  const long long feat = (long long)n_nodes * D;

  // in-degree -> 1/max(deg,1); recomputed every call (graph-replay safe)
  fill_zero_kernel<<<1024, 256, 0, stream>>>(invdeg, (long long)n_nodes);
  deg_count_kernel<<<(n_edges + 255) / 256, 256, 0, stream>>>(dst, invdeg, n_edges);
  deg_inv_kernel<<<(n_nodes + 255) / 256, 256, 0, stream>>>(invdeg, n_nodes);

  const float* cur = x0;
  for (int l = 0; l < n_layers; ++l) {
    fill_zero_kernel<<<4096, 256, 0, stream>>>(agg, feat);

    long long st = (long long)n_edges * 32;
    scatter_add_kernel<<<(int)((st + 255) / 256), 256, 0, stream>>>(
        cur, src, dst, agg, n_edges);

    // alternate targets so the final layer lands in d_out
    float* nxt = (((n_layers - 1 - l) & 1) == 0) ? outF : bufA;

    sage_layer_kernel<<<n_nodes / 16, 256, 0, stream>>>(
        cur, agg, invdeg,
        Wl + (size_t)l * D * D, bl + (size_t)l * D, Wr + (size_t)l * D * D,
        nxt);

    cur = nxt;
  }
}